// DPPDynamicEmbedding_79096117723300
// MI455X (gfx1250) — compile-verified
//
#include <hip/hip_runtime.h>

// MI455X / gfx1250, wave32.
// density via V_WMMA_F32_16X16X4_F32: d2 tile = A(16x4) x B(4x16) with
//   A row i = [x_i, y_i, sq_i, 1],  B col j = [-2x_j, -2y_j, 1, sq_j + (placed? 0 : 1e30)]
// Non-placed columns get +1e30 folded into d2, so a single f32 compare per
// 32 elements does the (within-radius AND placed) test. Counts accumulate on
// the dual-issue VALU; one xor-butterfly reduction at the end.

typedef __attribute__((ext_vector_type(2))) float v2f;
typedef __attribute__((ext_vector_type(8))) float v8f;

#define BDIM 16
#define NDIM 4096
#define EDIM 128
#define R2 (0.4f * 0.4f)
#define INV_SCALE (1.0f / 20.0f)
#define BIGOFF 1.0e30f

__device__ __forceinline__ v8f wmma_f32_16x16x4(v2f a, v2f b, v8f c) {
#if defined(__HIP_DEVICE_COMPILE__)
  return __builtin_amdgcn_wmma_f32_16x16x4_f32(
      /*neg_a=*/false, a, /*neg_b=*/false, b,
      /*c_mod=*/(short)0, c, /*reuse_a=*/false, /*reuse_b=*/false);
#else
  return c;  // host stub; never codegen'd for device
#endif
}

__global__ __launch_bounds__(32) void dpp_density_proj_kernel(
    const int* __restrict__ am,     // action_mask [B,N] (nonzero = still available)
    const int* __restrict__ ko,     // keepout     [B,N]
    const int* __restrict__ pr,     // probe       [B,N]
    const float* __restrict__ locs, // [B,N,2]
    const float* __restrict__ W,    // [2, 3E] row-major
    float* __restrict__ out)        // gk | gv | lk, each [B,N,E]
{
  const int lane = threadIdx.x;       // 0..31
  const int col  = lane & 15;         // column / row index inside 16-tile
  const int blk  = blockIdx.x;        // B * (N/16) blocks
  const int b    = blk >> 8;          // N/16 == 256 tiles per batch
  const int ibase = (blk & 255) << 4;
  const int base  = b * NDIM;
  const bool lowhalf = (lane < 16);

  // ---- A operand (16x4 f32): lanes 0-15 hold K0,K1; lanes 16-31 hold K2,K3
  const float2 pi = ((const float2*)locs)[base + ibase + col];
  const float sqi = pi.x * pi.x + pi.y * pi.y;
  v2f a;
  a.x = lowhalf ? pi.x : sqi;   // VGPR0: K0 | K2
  a.y = lowhalf ? pi.y : 1.0f;  // VGPR1: K1 | K3

  // per-lane partial counts: acc[k] = hits for (row k | k+8, columns == col mod 16)
  unsigned acc[8] = {0u, 0u, 0u, 0u, 0u, 0u, 0u, 0u};

#pragma unroll 2
  for (int jt = 0; jt < NDIM; jt += 16) {
    const int j = base + jt + col;
    const float2 pj2 = ((const float2*)locs)[j];
    const float sqj = pj2.x * pj2.x + pj2.y * pj2.y;

    // placed[j] = !available && !keepout && !probe; fold !placed into d2 offset
    const bool placed_j = ((am[j] | ko[j] | pr[j]) == 0);
    const float sqj_m = sqj + (placed_j ? 0.0f : BIGOFF);

    // B operand (4x16 f32): lanes 0-15 hold K0,K1; lanes 16-31 hold K2,K3
    v2f bm;
    bm.x = lowhalf ? (-2.0f * pj2.x) : 1.0f;   // K0 | K2
    bm.y = lowhalf ? (-2.0f * pj2.y) : sqj_m;  // K1 | K3

    v8f c = {0.f, 0.f, 0.f, 0.f, 0.f, 0.f, 0.f, 0.f};
    c = wmma_f32_16x16x4(a, bm, c);

    // c[k]: lanes 0-15 -> (row k, col lane); lanes 16-31 -> (row k+8, col lane-16)
#pragma unroll
    for (int k = 0; k < 8; ++k) acc[k] += (unsigned)(c[k] < R2);
  }

  // ---- reduce columns: xor-butterfly within each 16-lane half, then swap halves
  float dlo[8];  // density rows 0..7   (valid in every lane)
  float dhi[8];  // density rows 8..15  (valid in every lane)
#pragma unroll
  for (int k = 0; k < 8; ++k) {
    unsigned v = acc[k];
    v += (unsigned)__shfl_xor((int)v, 1, 32);
    v += (unsigned)__shfl_xor((int)v, 2, 32);
    v += (unsigned)__shfl_xor((int)v, 4, 32);
    v += (unsigned)__shfl_xor((int)v, 8, 32);
    const unsigned o = (unsigned)__shfl_xor((int)v, 16, 32);
    const unsigned lo = lowhalf ? v : o;
    const unsigned hi = lowhalf ? o : v;
    dlo[k] = (float)lo * INV_SCALE;
    dhi[k] = (float)hi * INV_SCALE;
  }

  // ---- projection + store: proj[n,e] = placed[n]*W[0,e] + density[n]*W[1,e]
  const size_t part = (size_t)BDIM * NDIM * EDIM;  // elements per output tensor
#pragma unroll
  for (int row = 0; row < 16; ++row) {
    const float dens = (row < 8) ? dlo[row] : dhi[row - 8];
    const int gidx = base + ibase + row;
    const float pl = ((am[gidx] | ko[gidx] | pr[gidx]) == 0) ? 1.0f : 0.0f;
    const size_t rowoff = (size_t)gidx * EDIM;
    for (int e = lane; e < 3 * EDIM; e += 32) {
      const float val = pl * W[e] + dens * W[3 * EDIM + e];
      const int seg = e >> 7;       // 0=glimpse_key 1=glimpse_val 2=logit_key
      const int ee  = e & (EDIM - 1);
      out[(size_t)seg * part + rowoff + ee] = val;
    }
  }
}

extern "C" void kernel_launch(void* const* d_in, const int* in_sizes, int n_in,
                              void* d_out, int out_size, void* d_ws, size_t ws_size,
                              hipStream_t stream) {
  (void)in_sizes; (void)n_in; (void)d_ws; (void)ws_size; (void)out_size;
  const int*   am   = (const int*)d_in[0];
  const int*   ko   = (const int*)d_in[1];
  const int*   pr   = (const int*)d_in[2];
  const float* locs = (const float*)d_in[3];
  const float* W    = (const float*)d_in[4];
  float* out = (float*)d_out;

  dim3 grid(BDIM * (NDIM / 16));  // 4096 waves, one 16-row i-tile each
  dpp_density_proj_kernel<<<grid, 32, 0, stream>>>(am, ko, pr, locs, W, out);
}